// project_C_dist_12610023981116
// MI455X (gfx1250) — compile-verified
//
#include <hip/hip_runtime.h>
#include <cstdint>
#include <cstddef>

// ---------------------------------------------------------------------------
// XPBD distance-constraint projection for MI455X (gfx1250, wave32).
//   outputs: d_out = [ V_new : 3*N_NODES floats | L_new : N_EDGES floats ]
// Memory-bound gather/scatter kernel: edge streams staged into LDS with the
// CDNA5 Tensor Data Mover (tensor_load_to_lds, TENSORcnt), node gathers served
// from the 192MB L2 (node data ~20MB), scatter via hardware f32 atomics.
// ---------------------------------------------------------------------------

#define TILE_EDGES       128           // edges staged per TDM tile (per wave)
#define WAVES_PER_BLOCK  8
#define BLOCK_THREADS    (WAVES_PER_BLOCK * 32)
#define BUF_BYTES        2048          // 128*8 (idx) + 128*4 (L) + 128*4 (rest)

typedef unsigned int u32x4 __attribute__((ext_vector_type(4)));
typedef int          i32x4 __attribute__((ext_vector_type(4)));
typedef int          i32x8 __attribute__((ext_vector_type(8)));

// Build a 1-D TDM descriptor (D#) and issue TENSOR_LOAD_TO_LDS.
// Layout per CDNA5 ISA:
//   group0: count=1 | lds_addr | global_addr[56:0] | type=2
//   group1: data_size=4B, tensor_dim0=elems_remain (tail reads OOB -> zero),
//           tensor_dim1=1, tile_dim0=tile_elems, tile_dim1/2=0 (1-D tile)
__device__ __forceinline__ void tdm_load_1d(uint32_t lds_addr, const void* gsrc,
                                            uint32_t elems_remain,
                                            uint32_t tile_elems) {
  uint64_t ga = (uint64_t)(uintptr_t)gsrc;
  u32x4 g0;
  g0[0] = 1u;                                   // count=1, is_restore=0, gather off
  g0[1] = lds_addr;                             // LDS byte address
  g0[2] = (uint32_t)ga;                         // global_addr[31:0]
  g0[3] = (uint32_t)(ga >> 32) | (2u << 30);    // global_addr[56:32] | type=2
  i32x8 g1;
  g1[0] = (int)(2u << 16);                      // workgroup_mask=0, data_size=2 (4B)
  g1[1] = (int)((elems_remain & 0xFFFFu) << 16);// tensor_dim0[15:0]
  g1[2] = (int)((elems_remain >> 16) | (1u << 16)); // tensor_dim0[31:16], tensor_dim1=1
  g1[3] = (int)(tile_elems << 16);              // tensor_dim1 hi=0, tile_dim0
  g1[4] = 0;                                    // tile_dim1=0, tile_dim2=0 (1-D)
  g1[5] = (int)elems_remain;                    // tensor_dim0_stride (unused for 1-D)
  g1[6] = 0;
  g1[7] = 0;
  i32x4 z4; z4[0] = z4[1] = z4[2] = z4[3] = 0;  // groups 2/3 unused (<=2-D tensor)
  i32x8 z8; z8[0] = z8[1] = z8[2] = z8[3] = 0;
  z8[4] = z8[5] = z8[6] = z8[7] = 0;
  // 6-arg toolchain variant: (g0, g1, g2, g3, extra_group, cpol)
  __builtin_amdgcn_tensor_load_to_lds(g0, g1, z4, z4, z8, /*cpol*/0);
}

// ---------------------------------------------------------------------------
// Kernel 1: V_new = V_predict  (vectorized stream copy, 24MB total)
// ---------------------------------------------------------------------------
__global__ void __launch_bounds__(256) copy_v_kernel(const float* __restrict__ src,
                                                     float* __restrict__ dst, int n) {
  int i4 = (blockIdx.x * blockDim.x + threadIdx.x) * 4;
  if (i4 + 3 < n) {
    *(float4*)(dst + i4) = *(const float4*)(src + i4);
  } else {
    for (; i4 < n; ++i4) dst[i4] = src[i4];
  }
}

// ---------------------------------------------------------------------------
// Kernel 2: per-edge constraint projection with TDM double-buffered staging.
// ---------------------------------------------------------------------------
__global__ void __launch_bounds__(BLOCK_THREADS)
xpbd_edge_kernel(const float* __restrict__ Vp,   // V_predict [3*nNodes]
                 const float* __restrict__ L,    // L        [nE]
                 const float* __restrict__ Vw,   // V_w      [nNodes]
                 const float* __restrict__ Vc,   // V_compliance [nNodes]
                 const int*   __restrict__ Cd,   // C_dist   [2*nE]
                 const float* __restrict__ R,    // C_init_d [nE]
                 float* __restrict__ Vout,       // V_new    [3*nNodes]
                 float* __restrict__ Lout,       // L_new    [nE]
                 int nE) {
  __shared__ __align__(16) unsigned char smem[WAVES_PER_BLOCK * 2 * BUF_BYTES];

  const int lane = threadIdx.x & 31;
  const int wave = __builtin_amdgcn_readfirstlane((int)(threadIdx.x >> 5));
  unsigned char* wbase = smem + (size_t)wave * (2 * BUF_BYTES);
  // Generic LDS pointers carry the LDS byte offset in their low 32 bits.
  const uint32_t ldsWave = (uint32_t)(uintptr_t)wbase;

  const int nTiles     = (nE + TILE_EDGES - 1) / TILE_EDGES;
  const int tileStride = (int)gridDim.x * WAVES_PER_BLOCK;
  int t = (int)blockIdx.x * WAVES_PER_BLOCK + wave;
  if (t >= nTiles) return;

  auto stage = [&](int tile, int buf) {
    const size_t base = (size_t)tile * TILE_EDGES;
    const uint32_t rem = (uint32_t)(nE - (int)base);   // >=1 by construction
    const uint32_t lb  = ldsWave + (uint32_t)buf * BUF_BYTES;
    tdm_load_1d(lb +    0, Cd + 2 * base, 2u * rem, 2u * TILE_EDGES); // indices
    tdm_load_1d(lb + 1024, L  +     base, rem,      TILE_EDGES);      // lambda
    tdm_load_1d(lb + 1536, R  +     base, rem,      TILE_EDGES);      // rest len
  };

  int buf = 0;
  stage(t, 0);                                   // prime the pipeline

  for (; t < nTiles; t += tileStride) {
    const int tn = t + tileStride;
    if (tn < nTiles) {
      stage(tn, buf ^ 1);                        // prefetch next tile (3 TDM ops)
      __builtin_amdgcn_s_wait_tensorcnt(3);      // current tile's 3 ops complete
    } else {
      __builtin_amdgcn_s_wait_tensorcnt(0);
    }
    asm volatile("" ::: "memory");               // keep LDS reads behind the wait

    const unsigned char* bb  = wbase + buf * BUF_BYTES;
    const int2*  idx = (const int2*) (bb);
    const float* lbp = (const float*)(bb + 1024);
    const float* rbp = (const float*)(bb + 1536);
    const int tbase = t * TILE_EDGES;

#pragma unroll
    for (int k = 0; k < TILE_EDGES / 32; ++k) {
      const int local = lane + 32 * k;           // conflict-free: stride-1 dwords
      const int e = tbase + local;
      if (e < nE) {
        const int2  ij = idx[local];
        const float lv = lbp[local];
        const float rv = rbp[local];
        const int i = ij.x, j = ij.y;

        // L2-resident gathers (node data ~20MB << 192MB L2)
        const float* pi = Vp + 3 * (size_t)i;
        const float* pj = Vp + 3 * (size_t)j;
        const float nx = pi[0] - pj[0];
        const float ny = pi[1] - pj[1];
        const float nz = pi[2] - pj[2];
        const float wi = Vw[i], wj = Vw[j];
        const float a  = 0.5f * (Vc[i] + Vc[j]);

        const float d  = sqrtf(nx * nx + ny * ny + nz * nz);
        const float c  = d - rv;
        float s = wi + wj;
        s = (s == 0.0f) ? __builtin_inff() : s;
        const float ld = (-c - a * lv) / (s + a);

        __builtin_nontemporal_store(lv + ld, Lout + e);  // streamed, never reread

        const float sc = ld / d;
        const float cx = nx * sc, cy = ny * sc, cz = nz * sc;

        float* oi = Vout + 3 * (size_t)i;
        float* oj = Vout + 3 * (size_t)j;
        unsafeAtomicAdd(oi + 0,  wi * cx);       // global_atomic_add_f32
        unsafeAtomicAdd(oi + 1,  wi * cy);
        unsafeAtomicAdd(oi + 2,  wi * cz);
        unsafeAtomicAdd(oj + 0, -wj * cx);
        unsafeAtomicAdd(oj + 1, -wj * cy);
        unsafeAtomicAdd(oj + 2, -wj * cz);
      }
    }
    buf ^= 1;
  }
}

// ---------------------------------------------------------------------------
extern "C" void kernel_launch(void* const* d_in, const int* in_sizes, int n_in,
                              void* d_out, int out_size, void* d_ws, size_t ws_size,
                              hipStream_t stream) {
  (void)n_in; (void)out_size; (void)d_ws; (void)ws_size;

  const float* Vp = (const float*)d_in[0];   // V_predict    [nNodes*3]
  const float* L  = (const float*)d_in[1];   // L            [nE]
  const float* Vw = (const float*)d_in[2];   // V_w          [nNodes]
  const float* Vc = (const float*)d_in[3];   // V_compliance [nNodes]
  const int*   Cd = (const int*)  d_in[4];   // C_dist       [nE*2]
  const float* R  = (const float*)d_in[5];   // C_init_d     [nE]

  const int nVf = in_sizes[0];               // 3 * nNodes
  const int nE  = in_sizes[1];               // nEdges

  float* Vout = (float*)d_out;               // first 3*nNodes floats
  float* Lout = Vout + nVf;                  // then nE floats

  // 1) V_new = V_predict
  {
    const int nThreads = (nVf + 3) / 4;
    const int blocks = (nThreads + 255) / 256;
    copy_v_kernel<<<blocks, 256, 0, stream>>>(Vp, Vout, nVf);
  }

  // 2) edge projection (TDM double-buffered; ~4 tiles per wave)
  {
    const int nTiles = (nE + TILE_EDGES - 1) / TILE_EDGES;
    int targetWaves = (nTiles + 3) / 4;
    int blocks = (targetWaves + WAVES_PER_BLOCK - 1) / WAVES_PER_BLOCK;
    if (blocks < 1) blocks = 1;
    xpbd_edge_kernel<<<blocks, BLOCK_THREADS, 0, stream>>>(
        Vp, L, Vw, Vc, Cd, R, Vout, Lout, nE);
  }
}